// BetweenClusterFC_88467736363908
// MI455X (gfx1250) — compile-verified
//
#include <hip/hip_runtime.h>
#include <hip/hip_bf16.h>

// ---------------------------------------------------------------------------
// BetweenClusterFC fused kernel for MI455X (gfx1250, wave32).
//   h1 = emb1 @ W1 + b1 ; h2 = emb2 @ W2 + b2 ; out = rowsum(h1 * h2)
// Native fp32 WMMA (V_WMMA_F32_16X16X4_F32) keeps full fp32 precision.
// ---------------------------------------------------------------------------

typedef float v2f __attribute__((ext_vector_type(2)));
typedef float v8f __attribute__((ext_vector_type(8)));

#define WMMA_F32(A, B, C)                                                  \
    __builtin_amdgcn_wmma_f32_16x16x4_f32(false, (A), false, (B),          \
                                          (short)0, (C), false, false)

namespace {
constexpr int D_IN   = 1024;   // K dimension
constexpr int F_OUT  = 512;    // per-matrix output features
constexpr int ROWS   = 32;     // rows per workgroup (2 M-tiles)
constexpr int KSLAB  = 128;    // K slab staged in LDS
constexpr int LSTR   = 132;    // padded LDS row stride (dwords) -> bank-conflict free
constexpr int BLOCK  = 512;    // 16 wave32 waves
constexpr int NWAVES = BLOCK / 32;
}  // namespace

__global__ __launch_bounds__(BLOCK) void between_cluster_fc_kernel(
    const float* __restrict__ e1, const float* __restrict__ e2,
    const float* __restrict__ W1, const float* __restrict__ b1,
    const float* __restrict__ W2, const float* __restrict__ b2,
    float* __restrict__ out) {
    __shared__ float sA1[ROWS * LSTR];
    __shared__ float sA2[ROWS * LSTR];
    __shared__ float sRed[ROWS * NWAVES];

    const int t    = threadIdx.x;
    const int w    = t >> 5;          // wave id 0..15
    const int lane = t & 31;
    const int half = lane >> 4;       // lane group: selects K-pair / row group
    const int lp   = lane & 15;       // position within 16-lane group
    const int r0   = blockIdx.x * ROWS;
    const int f0   = w * 32;          // this wave's 2 F-tiles: f0, f0+16

    // Per-lane bias values for the two F-tiles (column = f0 + 16*j + lp).
    const float bb1[2] = {b1[f0 + lp], b1[f0 + 16 + lp]};
    const float bb2[2] = {b2[f0 + lp], b2[f0 + 16 + lp]};

    // Accumulators: C{1,2}[m_tile][f_tile] -> 8 fp32 16x16 tiles (64 VGPRs).
    v8f C1[2][2] = {};
    v8f C2[2][2] = {};

    for (int ks = 0; ks < D_IN / KSLAB; ++ks) {
        const int k0 = ks * KSLAB;
        __syncthreads();  // previous slab fully consumed before overwrite

        // ---- stage 32 x 128 fp32 slabs of emb1/emb2 into LDS (B128 loads) --
#pragma unroll
        for (int c = 0; c < 2; ++c) {
            const int li  = (t + c * BLOCK) * 4;  // 0..4092 dwords
            const int row = li >> 7;              // / KSLAB
            const int kk  = li & (KSLAB - 1);
            const size_t goff = (size_t)(r0 + row) * D_IN + k0 + kk;
            const float4 v1 = *reinterpret_cast<const float4*>(e1 + goff);
            const float4 v2 = *reinterpret_cast<const float4*>(e2 + goff);
            *reinterpret_cast<float4*>(sA1 + row * LSTR + kk) = v1;
            *reinterpret_cast<float4*>(sA2 + row * LSTR + kk) = v2;
            if (ks + 1 < D_IN / KSLAB) {  // warm L2/WGP$ for the next slab
                __builtin_prefetch(e1 + goff + KSLAB, 0, 0);
                __builtin_prefetch(e2 + goff + KSLAB, 0, 0);
            }
        }
        __syncthreads();

        // ---- compute: 32 WMMA K-steps over the slab ------------------------
        for (int kl = 0; kl < KSLAB; kl += 4) {
            const int kw = kl + 2 * half;  // lane's K offset within slab
            // A tiles (16x4 fp32): lane = m + 16*(k>>1), vgpr = k&1.
            const v2f a1m0 = *reinterpret_cast<const v2f*>(sA1 + lp * LSTR + kw);
            const v2f a1m1 = *reinterpret_cast<const v2f*>(sA1 + (16 + lp) * LSTR + kw);
            const v2f a2m0 = *reinterpret_cast<const v2f*>(sA2 + lp * LSTR + kw);
            const v2f a2m1 = *reinterpret_cast<const v2f*>(sA2 + (16 + lp) * LSTR + kw);

            // B tiles (4x16 fp32): lane = n + 16*(k>>1), vgpr = k&1.
            const size_t kg  = (size_t)(k0 + kl + 2 * half);
            const float* w1p = W1 + kg * F_OUT + f0 + lp;
            const float* w2p = W2 + kg * F_OUT + f0 + lp;
            v2f bW1[2], bW2[2];
            bW1[0][0] = w1p[0];  bW1[0][1] = w1p[F_OUT];
            bW1[1][0] = w1p[16]; bW1[1][1] = w1p[F_OUT + 16];
            bW2[0][0] = w2p[0];  bW2[0][1] = w2p[F_OUT];
            bW2[1][0] = w2p[16]; bW2[1][1] = w2p[F_OUT + 16];

            C1[0][0] = WMMA_F32(a1m0, bW1[0], C1[0][0]);
            C1[0][1] = WMMA_F32(a1m0, bW1[1], C1[0][1]);
            C1[1][0] = WMMA_F32(a1m1, bW1[0], C1[1][0]);
            C1[1][1] = WMMA_F32(a1m1, bW1[1], C1[1][1]);
            C2[0][0] = WMMA_F32(a2m0, bW2[0], C2[0][0]);
            C2[0][1] = WMMA_F32(a2m0, bW2[1], C2[0][1]);
            C2[1][0] = WMMA_F32(a2m1, bW2[0], C2[1][0]);
            C2[1][1] = WMMA_F32(a2m1, bW2[1], C2[1][1]);
        }
    }

    // ---- epilogue: biases, elementwise product, partial F reduction --------
    // C/D layout: vgpr v, lane -> element (row = v + 8*half, col = lp).
    v8f P[2] = {};
#pragma unroll
    for (int m = 0; m < 2; ++m)
#pragma unroll
        for (int j = 0; j < 2; ++j)
#pragma unroll
            for (int v = 0; v < 8; ++v)
                P[m][v] += (C1[m][j][v] + bb1[j]) * (C2[m][j][v] + bb2[j]);

    // Horizontal sum over the 16 columns held across lanes (xor shuffles stay
    // inside each 16-lane group for masks 1,2,4,8), then cross-wave via LDS.
#pragma unroll
    for (int m = 0; m < 2; ++m)
#pragma unroll
        for (int v = 0; v < 8; ++v) {
            float s = P[m][v];
            s += __shfl_xor(s, 1);
            s += __shfl_xor(s, 2);
            s += __shfl_xor(s, 4);
            s += __shfl_xor(s, 8);
            if (lp == 0) {
                const int row = 16 * m + v + 8 * half;
                sRed[row * NWAVES + w] = s;
            }
        }
    __syncthreads();

    if (t < ROWS) {
        float acc = 0.0f;
#pragma unroll
        for (int ww = 0; ww < NWAVES; ++ww) acc += sRed[t * NWAVES + ww];
        out[r0 + t] = acc;
    }
}

extern "C" void kernel_launch(void* const* d_in, const int* in_sizes, int n_in,
                              void* d_out, int out_size, void* d_ws, size_t ws_size,
                              hipStream_t stream) {
    const float* e1 = (const float*)d_in[0];
    const float* e2 = (const float*)d_in[1];
    const float* W1 = (const float*)d_in[2];
    const float* b1 = (const float*)d_in[3];
    const float* W2 = (const float*)d_in[4];
    const float* b2 = (const float*)d_in[5];
    // d_in[6], d_in[7] are cluster ids; weights are already the selected pair.
    float* out = (float*)d_out;

    const int n = in_sizes[0] / D_IN;  // 32768
    dim3 grid(n / ROWS), block(BLOCK);
    between_cluster_fc_kernel<<<grid, block, 0, stream>>>(e1, e2, W1, b1, W2, b2, out);
}